// AdderLayerCUDA_16209206575290
// MI455X (gfx1250) — compile-verified
//
#include <hip/hip_runtime.h>
#include <math.h>

#define KTAPS 9
#define CCH 64
#define OCH 64
#define HH 64
#define WW 64
#define NBUF 2
#define OBLK 32   // output channels per block
#define CB 4      // channels staged per pipeline step

// ---- CDNA5 async Global->LDS copy (ASYNCcnt-tracked) -----------------------
// GVS addressing: mem_addr = SADDR(64b sgpr pair) + VADDR(32b unsigned vgpr).
// VDST holds the LDS byte address (low 32 bits of the generic shared pointer).
__device__ __forceinline__ void async_ld_b32(uint32_t lds_addr, uint32_t voff,
                                             const float* sbase) {
    asm volatile("global_load_async_to_lds_b32 %0, %1, %2"
                 :
                 : "v"(lds_addr), "v"(voff), "s"(sbase)
                 : "memory");
}
// Async loads complete in order: after issuing an 8-wide batch, waiting
// ASYNCcnt<=8 guarantees the *previous* 8-wide batch has fully landed in LDS.
__device__ __forceinline__ void async_wait8() {
    asm volatile("s_wait_asynccnt 8" ::: "memory");
}
__device__ __forceinline__ void async_wait0() {
    asm volatile("s_wait_asynccnt 0" ::: "memory");
}

__global__ __launch_bounds__(256) void adder2d_pow_kernel(
    const float* __restrict__ x, const float* __restrict__ w,
    const float* __restrict__ alpha, float* __restrict__ out) {
    // Double-buffered staging of CB channels: 3 halo rows of x (66 cols) and a
    // 32-o weight slice per channel.  Total ~15.6 KB of the 320 KB WGP LDS.
    __shared__ float xrow[NBUF][CB][3][WW + 2];   // 2*4*792 B
    __shared__ float wsl[NBUF][CB][OBLK * KTAPS]; // 2*4*1152 B
    __shared__ float dummy[16];                   // sink for padded slots

    const int t  = threadIdx.x;   // 0..255
    const int j  = t & (WW - 1);  // output column
    const int og = t >> 6;        // 0..3 -> 8 output channels each

    const int bi   = blockIdx.x;      // B*H*2 = 1024 blocks
    const int half = bi & 1;          // which 32-o half
    const int rb   = bi >> 1;
    const int b    = rb >> 6;         // batch
    const int i    = rb & 63;         // output row

    // Zero both x staging buffers once: provides zero padding at W borders and
    // for out-of-range halo rows (invalid slots are redirected to `dummy` with
    // zero LDS stride, so the zeroed rows are never overwritten).
    for (int k = t; k < NBUF * CB * 3 * (WW + 2); k += 256)
        ((float*)xrow)[k] = 0.0f;
    __syncthreads();

    float acc[8];
#pragma unroll
    for (int oo = 0; oo < 8; ++oo) acc[oo] = 0.0f;

    // ---- per-thread async assignments: EXACTLY 2 loads per thread/channel --
    // Unified slot space u = t + q*256, u in [0,512):
    //   u <  192 : x halo dword   (row u/64, col u%64)
    //   u >= 192 : weight dword d = u-192 (valid d < 288, else dummy sink)
    // Validity boundaries fall on wave32 boundaries, so per-wave ASYNCcnt
    // increments are exactly 2*CB per staged tile -> s_wait_asynccnt 8 is an
    // exact double-buffer wait.
    uint32_t ldsA[2][NBUF]; // LDS byte address (channel 0 of tile) per buffer
    uint32_t ldst[2];       // LDS byte stride per channel within a tile
    uint32_t off0[2];       // global byte offset at c=0
    uint32_t cstp[2];       // global byte stride per channel
    bool     isx[2];
#pragma unroll
    for (int q = 0; q < 2; ++q) {
        const int u = t + q * 256;
        if (u < 192) {
            isx[q] = true;
            const int  r   = u >> 6;        // 0..2 halo row
            const int  col = u & 63;
            const int  gi  = i - 1 + r;     // global input row
            const bool v   = (gi >= 0) && (gi < HH);
            const int  gic = v ? gi : 0;    // clamped (address stays valid)
            off0[q] = (uint32_t)(((b * CCH * HH + gic) * WW + col) * 4);
            cstp[q] = (uint32_t)(HH * WW * 4);
            ldst[q] = v ? (uint32_t)(3 * (WW + 2) * 4) : 0u;
#pragma unroll
            for (int nb = 0; nb < NBUF; ++nb)
                ldsA[q][nb] = (uint32_t)(uintptr_t)(v ? &xrow[nb][0][r][1 + col]
                                                      : &dummy[t & 15]);
        } else {
            isx[q] = false;
            const int  d  = u - 192;
            const bool v  = (d < OBLK * KTAPS);
            const int  dd = v ? d : 0;
            const int  ol  = dd / 9;             // local o: 0..31
            const int  tap = dd % 9;
            const int  o   = half * OBLK + ol;
            off0[q] = (uint32_t)(((o * CCH) * KTAPS + tap) * 4);
            cstp[q] = (uint32_t)(KTAPS * 4);
            ldst[q] = v ? (uint32_t)(OBLK * KTAPS * 4) : 0u;
#pragma unroll
            for (int nb = 0; nb < NBUF; ++nb)
                ldsA[q][nb] = (uint32_t)(uintptr_t)(v ? &wsl[nb][0][dd]
                                                      : &dummy[t & 15]);
        }
    }

    // Stage CB consecutive channels (tile `ct`) into buffer `nb`.
    auto issue_tile = [&](int ct, int nb) {
#pragma unroll
        for (int ci = 0; ci < CB; ++ci) {
            const uint32_t c = (uint32_t)(ct * CB + ci);
#pragma unroll
            for (int q = 0; q < 2; ++q) {
                const uint32_t lds  = ldsA[q][nb] + (uint32_t)ci * ldst[q];
                const uint32_t voff = off0[q] + c * cstp[q];
                if (isx[q]) async_ld_b32(lds, voff, x);
                else        async_ld_b32(lds, voff, w);
            }
        }
    };

    // Prologue: stage tile 0 into buffer 0.
    issue_tile(0, 0);

    const int NT = CCH / CB; // 16 channel tiles
    for (int ct = 0; ct < NT; ++ct) {
        const int cur = ct & 1;
        if (ct + 1 < NT) {
            issue_tile(ct + 1, cur ^ 1);
            async_wait8(); // tile ct landed; tile ct+1 may still be in flight
        } else {
            async_wait0();
        }
        __syncthreads();

        // ---- compute: CB channels x 9 taps x 8 output channels ----
#pragma unroll
        for (int ci = 0; ci < CB; ++ci) {
            float xv[9];
#pragma unroll
            for (int r = 0; r < 3; ++r)
#pragma unroll
                for (int cc = 0; cc < 3; ++cc)
                    xv[r * 3 + cc] = xrow[cur][ci][r][j + cc]; // center j+1

#pragma unroll
            for (int oo = 0; oo < 8; ++oo) {
                const float* wp = &wsl[cur][ci][(og * 8 + oo) * KTAPS];
                float s = acc[oo];
#pragma unroll
                for (int u = 0; u < KTAPS; ++u) s += fabsf(xv[u] - wp[u]);
                acc[oo] = s;
            }
        }
        __syncthreads(); // all waves done reading `cur` before it is re-staged
    }

    // ---- epilogue: residual add + sign(y)*|y|^alpha ----
    const float alphaV = alpha[0];
#pragma unroll
    for (int oo = 0; oo < 8; ++oo) {
        const int o   = half * OBLK + og * 8 + oo;
        const int idx = (((b * OCH) + o) * HH + i) * WW + j;
        float y = x[idx] - acc[oo]; // O == C, residual uses channel o
        float r = powf(fabsf(y), alphaV);
        out[idx] = (y > 0.0f) ? r : ((y < 0.0f) ? -r : 0.0f);
    }
}

extern "C" void kernel_launch(void* const* d_in, const int* in_sizes, int n_in,
                              void* d_out, int out_size, void* d_ws, size_t ws_size,
                              hipStream_t stream) {
    const float* x     = (const float*)d_in[0];
    const float* w     = (const float*)d_in[1];
    const float* alpha = (const float*)d_in[2];
    float*       out   = (float*)d_out;

    dim3 grid(8 * HH * 2); // B * H * 2 = 1024 blocks
    dim3 block(256);       // 8 wave32s
    adder2d_pow_kernel<<<grid, block, 0, stream>>>(x, w, alpha, out);
}